// ChebySemi_55645596286988
// MI455X (gfx1250) — compile-verified
//
#include <hip/hip_runtime.h>
#include <math.h>
#include <stdint.h>

#ifndef M_PI
#define M_PI 3.14159265358979323846
#endif

// Problem constants (from the reference)
#define IMG_H 512
#define IMG_W 512
#define NBATCH 64
#define POWER_ITERS 20
#define NITERS 15

// Block computes a 32x32 output region (4 waves x one 16x16 WMMA tile each).
// Halo tile 34x34 stored densely (stride 34 dwords: 34=2*17, odd co-factor ->
// conflict-free b64 banking). K-pad reads wrap into the next row (finite
// data x B==0); 2 zeroed tail dwords cover the wrap of the last row.
#define HALO 34
#define LSTRIDE 34
#define LDS_ELEMS (HALO * HALO + 2)
#define NCHUNK 15   // 3 kernel rows x 5 K-chunks (K=18 padded to 20, K_wmma=4)

typedef float v2f __attribute__((ext_vector_type(2)));
typedef float v8f __attribute__((ext_vector_type(8)));
typedef unsigned int v4u __attribute__((ext_vector_type(4)));
typedef int v4i __attribute__((ext_vector_type(4)));
typedef int v8i __attribute__((ext_vector_type(8)));

// ---------------------------------------------------------------------------
// One-shot per-batch Toeplitz fragment build (WMMA B-operand register layout):
//   chunk = dr*5+kc;  lane holds B[2*hi+v][lo], hi=lane>>4, lo=lane&15
//   B[c][n] = K[dr][c-n] if 0<=c-n<=2 else 0,   c = kc*4 + 2*hi + v
// ---------------------------------------------------------------------------
__global__ void build_toeplitz(const float* __restrict__ kern,
                               v2f* __restrict__ bmat) {
    int gid = blockIdx.x * blockDim.x + threadIdx.x;
    if (gid >= NBATCH * NCHUNK * 32) return;
    int lane  = gid & 31;
    int chunk = (gid >> 5) % NCHUNK;
    int b     = gid / (NCHUNK * 32);
    int dr = chunk / 5, kc = chunk - dr * 5;
    int hi = lane >> 4, lo = lane & 15;
    v2f bv;
#pragma unroll
    for (int v = 0; v < 2; ++v) {
        int c = kc * 4 + 2 * hi + v;
        int idx = c - lo;
        float val = (idx >= 0 && idx <= 2) ? kern[b * 9 + dr * 3 + idx] : 0.0f;
        if (v == 0) bv.x = val; else bv.y = val;
    }
    bmat[gid] = bv;
}

// ---------------------------------------------------------------------------
// Interior tile load: one Tensor Data Mover descriptor per block.
// D# per CDNA5 ISA ch8: 2-D tensor (512x512 f32, row stride 512), 34x34 tile,
// tail descriptor groups zero/null. Wave 0 issues TENSOR_LOAD_TO_LDS and
// retires it on TENSORcnt before the block barrier.
// ---------------------------------------------------------------------------
__device__ __forceinline__ void load_halo_tdm(float* __restrict__ lds,
                                              const float* __restrict__ img,
                                              int r0, int c0, int tid) {
#if __has_builtin(__builtin_amdgcn_tensor_load_to_lds)
    // zero the 2 wrap tail dwords (disjoint from the TDM tile region)
    if (tid < 2) lds[HALO * HALO + tid] = 0.0f;
    if (tid < 32) {
        unsigned lds_addr = (unsigned)(uintptr_t)lds;
        unsigned long long ga =
            (unsigned long long)(uintptr_t)(img + (r0 - 1) * IMG_W + (c0 - 1));
        v4u g0 = { 1u,                                   // count=1 (valid user D#)
                   lds_addr,                             // lds_addr (bytes)
                   (unsigned)ga,                         // global_addr[31:0]
                   ((unsigned)(ga >> 32) & 0x01FFFFFFu)  // global_addr[56:32]
                       | 0x80000000u };                  // type=2 ("image")
        v8i g1 = { 0x20000,                 // data_size=2 (4B); mask/flags=0
                   (int)(512u << 16),       // tensor_dim0[15:0] @bit48
                   (int)(512u << 16),       // tensor_dim1[15:0] @bit80
                   (int)(34u  << 16),       // tile_dim0 @bit112
                   34,                      // tile_dim1 @bit128 (tile_dim2=0)
                   512,                     // tensor_dim0_stride[31:0] @bit160
                   0, 0 };
        v4i g2 = { 0, 0, 0, 0 };
        v4i g3 = { 0, 0, 0, 0 };
        v8i g4 = { 0, 0, 0, 0, 0, 0, 0, 0 };   // clang-23 6-arg form tail group
        __builtin_amdgcn_tensor_load_to_lds(g0, g1, g2, g3, g4, 0);
        __builtin_amdgcn_s_wait_tensorcnt(0);
    }
#else
    // fallback: per-lane async copy of the whole (in-bounds) tile
    const float* base = img + (r0 - 1) * IMG_W + (c0 - 1);
    unsigned lds_base = (unsigned)(uintptr_t)lds;
    for (int idx = tid; idx < HALO * HALO; idx += 128) {
        int r = idx / HALO, c = idx - r * HALO;
        asm volatile("global_load_async_to_lds_b32 %0, %1, %2"
                     :: "v"(lds_base + (unsigned)idx * 4u),
                        "v"((r * IMG_W + c) * 4), "s"(base) : "memory");
    }
    if (tid < 2) lds[HALO * HALO + tid] = 0.0f;
    asm volatile("s_wait_asynccnt 0" ::: "memory");
#endif
}

// ---------------------------------------------------------------------------
// Boundary tile load: zero-fill LDS, barrier (commits DS stores), then
// EXEC-masked CDNA5 async copies for the in-bounds elements only.
// ---------------------------------------------------------------------------
__device__ __forceinline__ void load_halo_boundary(float* __restrict__ lds,
                                                   const float* __restrict__ img,
                                                   int r0, int c0, int tid) {
    for (int idx = tid; idx < LDS_ELEMS; idx += 128) lds[idx] = 0.0f;
    __syncthreads();   // DS stores visible before async writes land
    unsigned lds_base = (unsigned)(uintptr_t)lds;
    for (int idx = tid; idx < HALO * HALO; idx += 128) {
        int r = idx / HALO, c = idx - r * HALO;
        int gr = r0 - 1 + r, gc = c0 - 1 + c;
        if (gr >= 0 && gr < IMG_H && gc >= 0 && gc < IMG_W) {
            asm volatile("global_load_async_to_lds_b32 %0, %1, %2"
                         :: "v"(lds_base + (unsigned)idx * 4u),
                            "v"((gr * IMG_W + gc) * 4), "s"(img) : "memory");
        }
    }
    asm volatile("s_wait_asynccnt 0" ::: "memory");
}

__device__ __forceinline__ void load_tile_lds(float* __restrict__ lds,
                                              const float* __restrict__ img,
                                              int r0, int c0, int tid) {
    // block-uniform branch -> EXEC stays full inside each path
    if (r0 > 0 && c0 > 0 && r0 < IMG_H - 32 && c0 < IMG_W - 32)
        load_halo_tdm(lds, img, r0, c0, tid);
    else
        load_halo_boundary(lds, img, r0, c0, tid);
    __syncthreads();
}

// ---------------------------------------------------------------------------
// 16x16 conv tile: O = sum_{dr} S_dr(16x18) * T_dr(18x16) via 15 chained
// accumulating V_WMMA_F32_16X16X4_F32.
// ---------------------------------------------------------------------------
__device__ __forceinline__ v8f conv_tile(const float* __restrict__ lds,
                                         const v2f* __restrict__ bm,
                                         int tr, int tc, int lane) {
    const int hi = lane >> 4;
    const int lo = lane & 15;
    v2f bfrag[NCHUNK];
#pragma unroll
    for (int ch = 0; ch < NCHUNK; ++ch)
        bfrag[ch] = bm[ch * 32 + lane];

    v8f acc = {};
#pragma unroll
    for (int dr = 0; dr < 3; ++dr) {
        const float* lrow = lds + (tr + lo + dr) * LSTRIDE;
#pragma unroll
        for (int kc = 0; kc < 5; ++kc) {
            int cb = tc + kc * 4 + 2 * hi;
            v2f a;
            a.x = lrow[cb];
            a.y = lrow[cb + 1];
            acc = __builtin_amdgcn_wmma_f32_16x16x4_f32(
                false, a, false, bfrag[dr * 5 + kc], (short)0, acc, false, false);
        }
    }
    return acc;
}

// ---------------------------------------------------------------------------
// Power iteration sweep: y = conv(src) * (1/m_prev); m_next[b] = max|y|.
// ---------------------------------------------------------------------------
__global__ __launch_bounds__(128) void conv_power(const float* __restrict__ src,
                                                  float* __restrict__ dst,
                                                  const v2f* __restrict__ bmat,
                                                  const float* __restrict__ m_prev,
                                                  unsigned* __restrict__ m_next) {
    __shared__ float lds[LDS_ELEMS];
    __shared__ float wred[4];

    const int bid   = blockIdx.x;
    const int batch = bid >> 8;           // 256 blocks per batch
    const int t     = bid & 255;
    const int r0    = (t >> 4) << 5;      // 32x32 region origin
    const int c0    = (t & 15) << 5;

    const float scale = __builtin_amdgcn_rcpf(m_prev[batch]);

    load_tile_lds(lds, src + (size_t)batch * IMG_H * IMG_W, r0, c0, threadIdx.x);

    const int wid  = threadIdx.x >> 5;
    const int lane = threadIdx.x & 31;
    const int tr = (wid >> 1) << 4;
    const int tc = (wid & 1)  << 4;

    v8f acc = conv_tile(lds, bmat + batch * (NCHUNK * 32), tr, tc, lane);

    const int hi = lane >> 4, lo = lane & 15;
    float* obase = dst + (size_t)batch * IMG_H * IMG_W;
    float amax = 0.0f;
#pragma unroll
    for (int v = 0; v < 8; ++v) {
        int orow = r0 + tr + v + 8 * hi;
        int ocol = c0 + tc + lo;
        float y = acc[v] * scale;
        obase[orow * IMG_W + ocol] = y;
        amax = fmaxf(amax, fabsf(y));
    }
#pragma unroll
    for (int off = 16; off > 0; off >>= 1)
        amax = fmaxf(amax, __shfl_xor(amax, off, 32));
    if (lane == 0) wred[wid] = amax;
    __syncthreads();
    if (threadIdx.x == 0) {
        float m = fmaxf(fmaxf(wred[0], wred[1]), fmaxf(wred[2], wred[3]));
        atomicMax(m_next + batch, __float_as_uint(m));  // bits-compare OK: m >= 0
    }
}

// ---------------------------------------------------------------------------
// Chebyshev sweep: xout = xin + tau * (f - conv(xin)),
//   tau = 2 / (m * (4/3 + (2/3)*root))
// ---------------------------------------------------------------------------
__global__ __launch_bounds__(128) void conv_cheby(const float* __restrict__ xin,
                                                  float* __restrict__ xout,
                                                  const float* __restrict__ f,
                                                  const v2f* __restrict__ bmat,
                                                  const float* __restrict__ m_final,
                                                  float root) {
    __shared__ float lds[LDS_ELEMS];

    const int bid   = blockIdx.x;
    const int batch = bid >> 8;
    const int t     = bid & 255;
    const int r0    = (t >> 4) << 5;
    const int c0    = (t & 15) << 5;

    const float m   = m_final[batch];
    const float tau =
        2.0f * __builtin_amdgcn_rcpf(m * (4.0f / 3.0f + (2.0f / 3.0f) * root));

    load_tile_lds(lds, xin + (size_t)batch * IMG_H * IMG_W, r0, c0, threadIdx.x);

    const int wid  = threadIdx.x >> 5;
    const int lane = threadIdx.x & 31;
    const int tr = (wid >> 1) << 4;
    const int tc = (wid & 1)  << 4;

    v8f acc = conv_tile(lds, bmat + batch * (NCHUNK * 32), tr, tc, lane);

    const int hi = lane >> 4, lo = lane & 15;
    const float* fbase = f    + (size_t)batch * IMG_H * IMG_W;
    float*       obase = xout + (size_t)batch * IMG_H * IMG_W;
#pragma unroll
    for (int v = 0; v < 8; ++v) {
        int m_out = v + 8 * hi;
        int orow  = r0 + tr + m_out;
        int ocol  = c0 + tc + lo;
        float xc = lds[(tr + m_out + 1) * LSTRIDE + (tc + lo + 1)];  // center x
        float fv = fbase[orow * IMG_W + ocol];
        obase[orow * IMG_W + ocol] = xc + tau * (fv - acc[v]);
    }
}

// ---------------------------------------------------------------------------
// m-array init: slot 0 = 1.0f, slots 1..20 = 0. Re-run every call.
// ---------------------------------------------------------------------------
__global__ void init_m(unsigned* __restrict__ m) {
    int i = threadIdx.x + blockIdx.x * blockDim.x;
    if (i < (POWER_ITERS + 1) * NBATCH)
        m[i] = (i < NBATCH) ? 0x3F800000u : 0u;
}

// ---------------------------------------------------------------------------
// Host: 1 init + 1 toeplitz-build + 20 power + 15 cheby launches.
// Ping-pong between d_ws buffer and d_out (~64.2 MB scratch total).
// ---------------------------------------------------------------------------
extern "C" void kernel_launch(void* const* d_in, const int* in_sizes, int n_in,
                              void* d_out, int out_size, void* d_ws, size_t ws_size,
                              hipStream_t stream) {
    const float* x    = (const float*)d_in[0];
    const float* f    = (const float*)d_in[1];
    const float* kern = (const float*)d_in[2];
    const float* u    = (const float*)d_in[3];
    float* out = (float*)d_out;

    unsigned* m_arr = (unsigned*)d_ws;                    // (20+1) x 64 u32
    v2f*  bmat = (v2f*)((float*)d_ws + 2048);             // 64 x 15 x 32 float2
    float* buf = (float*)d_ws + 32768;                    // 64 MB ping-pong

    const dim3 grid(NBATCH * 256), block(128);

    init_m<<<dim3(6), dim3(256), 0, stream>>>(m_arr);
    build_toeplitz<<<dim3((NBATCH * NCHUNK * 32 + 255) / 256), dim3(256), 0, stream>>>(
        kern, bmat);

    // Power iterations: even sweeps -> ws, odd -> out.
    const float* src = u;
    for (int t = 0; t < POWER_ITERS; ++t) {
        float* dst = (t & 1) ? out : buf;
        conv_power<<<grid, block, 0, stream>>>(
            src, dst, bmat,
            (const float*)(m_arr + t * NBATCH), m_arr + (t + 1) * NBATCH);
        src = dst;
    }

    // Chebyshev sweeps: even -> out, odd -> ws; step 14 (even) lands in d_out.
    const float* mf = (const float*)(m_arr + POWER_ITERS * NBATCH);
    const float* xs = x;
    for (int i = 0; i < NITERS; ++i) {
        float root = (float)cos(M_PI * (2 * i + 1) / (2.0 * NITERS));
        float* dst = (i & 1) ? buf : out;
        conv_cheby<<<grid, block, 0, stream>>>(xs, dst, f, bmat, mf, root);
        xs = dst;
    }
}